// GraphEncoder_23914377904254
// MI455X (gfx1250) — compile-verified
//
#include <hip/hip_runtime.h>
#include <hip/hip_bf16.h>

#define NN 100000
#define EE 3200000
#define NODE_IN 118
#define DD 32
#define HH 4
#define CC 8
#define EDIM 53
#define EDIM_PAD 56
#define KIN_PAD 120
#define LL 4

typedef __attribute__((ext_vector_type(2))) float v2f;
typedef __attribute__((ext_vector_type(8))) float v8f;

// ---------------- WMMA helpers (fp32, 16x16x4) ----------------
// A: 16x4 f32. Lane L: M = L&15, VGPR0 holds K = 2*(L>>4), VGPR1 holds K+1.
// B: 4x16 f32. Lane L: N = L&15, VGPR0 holds K = 2*(L>>4), VGPR1 holds K+1.
// C/D: 16x16 f32. Lane L, VGPR r: M = r + 8*(L>>4), N = L&15.

__device__ __forceinline__ v8f wmma4(v2f a, v2f b, v8f c) {
  return __builtin_amdgcn_wmma_f32_16x16x4_f32(false, a, false, b, (short)0, c,
                                               false, false);
}

__device__ __forceinline__ v2f afrag(const float* tile, int stride, int lane,
                                     int k0) {
  int m = lane & 15;
  int kb = (lane >> 4) << 1;
  v2f a;
  a.x = tile[m * stride + k0 + kb];
  a.y = tile[m * stride + k0 + kb + 1];
  return a;
}

// B fragment from LDS weight staged in K-pair-interleaved, bank-swizzled
// layout: wls[p*64 + (n ^ ((p&1)<<4))*2 + j] = W[2p+j][n]  (zero-padded).
// One aligned ds_load_b64 per fragment, no register shuffles.
__device__ __forceinline__ v2f bfragP(const float* bW, int lane, int k0,
                                      int nt) {
  int n = (lane & 15) + nt * 16;
  int p = (k0 >> 1) + (lane >> 4);
  int slot = n ^ ((p & 1) << 4);
  return *(const v2f*)(bW + p * 64 + slot * 2);
}

// stage a [Kdim][32] row-major weight into the interleaved layout, padded to
// Kpad (Kpad even). `nthreads` = blockDim.x.
__device__ __forceinline__ void stageW(const float* __restrict__ W, float* wls,
                                       int Kdim, int Kpad) {
  for (int idx = threadIdx.x; idx < Kpad * DD; idx += 128) {
    int p = idx >> 6;
    int r = idx & 63;
    int slot = r >> 1, j = r & 1;
    int n = slot ^ ((p & 1) << 4);
    int k = 2 * p + j;
    wls[idx] = (k < Kdim) ? W[k * DD + n] : 0.f;
  }
}

// monotonic uint encoding of float for atomicMax-based segment max
__device__ __forceinline__ unsigned fkey(float f) {
  unsigned u = __float_as_uint(f);
  return (u & 0x80000000u) ? ~u : (u | 0x80000000u);
}
__device__ __forceinline__ float fdec(unsigned k) {
  return (k & 0x80000000u) ? __uint_as_float(k ^ 0x80000000u)
                           : __uint_as_float(~k);
}

// ---------------- input projection: h = x @ W + b ----------------
__global__ void k_input_proj(const float* __restrict__ x,
                             const float* __restrict__ W,
                             const float* __restrict__ bias,
                             float* __restrict__ h) {
  __shared__ float wls[KIN_PAD * DD];
  __shared__ float xls[4][16 * KIN_PAD];
  int wave = threadIdx.x >> 5, lane = threadIdx.x & 31;
  int base = (blockIdx.x * 4 + wave) * 16;
  stageW(W, wls, NODE_IN, KIN_PAD);
  float* t = xls[wave];
  for (int idx = lane; idx < 16 * KIN_PAD; idx += 32) {
    int r = idx / KIN_PAD, c = idx - r * KIN_PAD;
    int node = base + r;
    t[idx] = (node < NN && c < NODE_IN) ? x[node * NODE_IN + c] : 0.f;
  }
  __syncthreads();
  v8f acc0 = {}, acc1 = {};
  for (int k0 = 0; k0 < KIN_PAD; k0 += 4) {
    v2f a = afrag(t, KIN_PAD, lane, k0);
    acc0 = wmma4(a, bfragP(wls, lane, k0, 0), acc0);
    acc1 = wmma4(a, bfragP(wls, lane, k0, 1), acc1);
  }
  int cb = lane & 15, rb = (lane >> 4) << 3;
  for (int r = 0; r < 8; ++r) {
    int node = base + rb + r;
    if (node < NN) {
      h[node * DD + cb] = acc0[r] + bias[cb];
      h[node * DD + 16 + cb] = acc1[r] + bias[16 + cb];
    }
  }
}

// ---------------- LN + 4 projections (q,k,v,skip) ----------------
__device__ __forceinline__ void proj16(const float* t, const float* bW,
                                       const float* __restrict__ bias,
                                       float* __restrict__ out, int lane,
                                       int base) {
  v8f acc0 = {}, acc1 = {};
  for (int k0 = 0; k0 < DD; k0 += 4) {
    v2f a = afrag(t, DD, lane, k0);
    acc0 = wmma4(a, bfragP(bW, lane, k0, 0), acc0);
    acc1 = wmma4(a, bfragP(bW, lane, k0, 1), acc1);
  }
  int cb = lane & 15, rb = (lane >> 4) << 3;
  for (int r = 0; r < 8; ++r) {
    int node = base + rb + r;
    if (node < NN) {
      out[node * DD + cb] = acc0[r] + bias[cb];
      out[node * DD + 16 + cb] = acc1[r] + bias[16 + cb];
    }
  }
}

__global__ void k_ln_proj(const float* __restrict__ h,
                          const float* __restrict__ g,
                          const float* __restrict__ bln,
                          const float* __restrict__ Wq, const float* __restrict__ bq,
                          const float* __restrict__ Wk, const float* __restrict__ bk,
                          const float* __restrict__ Wv, const float* __restrict__ bv,
                          const float* __restrict__ Ws, const float* __restrict__ bs,
                          float* __restrict__ qbuf, float* __restrict__ kbuf,
                          float* __restrict__ vbuf, float* __restrict__ sbuf) {
  __shared__ float wls[4][DD * DD];  // q,k,v,skip weights (interleaved)
  __shared__ float xls[4][16 * DD];
  int wave = threadIdx.x >> 5, lane = threadIdx.x & 31;
  int base = (blockIdx.x * 4 + wave) * 16;
  stageW(Wq, wls[0], DD, DD);
  stageW(Wk, wls[1], DD, DD);
  stageW(Wv, wls[2], DD, DD);
  stageW(Ws, wls[3], DD, DD);
  float* t = xls[wave];
  for (int idx = lane; idx < 16 * DD; idx += 32) {
    int r = idx >> 5, c = idx & 31;
    int node = base + r;
    t[idx] = (node < NN) ? h[node * DD + c] : 0.f;
  }
  __syncthreads();
  if (lane < 16) {
    float* row = t + lane * DD;
    float mu = 0.f;
    for (int c = 0; c < DD; ++c) mu += row[c];
    mu *= (1.f / DD);
    float var = 0.f;
    for (int c = 0; c < DD; ++c) {
      float d = row[c] - mu;
      var += d * d;
    }
    var *= (1.f / DD);
    float inv = rsqrtf(var + 1e-5f);
    for (int c = 0; c < DD; ++c) row[c] = (row[c] - mu) * inv * g[c] + bln[c];
  }
  __syncthreads();
  proj16(t, wls[0], bq, qbuf, lane, base);
  proj16(t, wls[1], bk, kbuf, lane, base);
  proj16(t, wls[2], bv, vbuf, lane, base);
  proj16(t, wls[3], bs, sbuf, lane, base);
}

// ------------- edge GEMM tile into LDS (shared by two edge passes) -------------
__device__ __forceinline__ void edge_gemm_tile(const float* __restrict__ elen,
                                               const float* __restrict__ evec,
                                               const float* wels, float* ef,
                                               float* et, int lane, int base) {
  const float2* elen2 = (const float2*)elen;
  for (int t = lane; t < 16 * 25; t += 32) {  // 50 cols as 25 float2 per row
    int r = t / 25, c2 = t - r * 25;
    int e = base + r;
    float2 v = make_float2(0.f, 0.f);
    if (e < EE) v = elen2[e * 25 + c2];
    ef[r * EDIM_PAD + c2 * 2] = v.x;
    ef[r * EDIM_PAD + c2 * 2 + 1] = v.y;
  }
  for (int t = lane; t < 16 * 3; t += 32) {  // edge_vec cols 50..52
    int r = t / 3, c = t - r * 3;
    int e = base + r;
    ef[r * EDIM_PAD + 50 + c] = (e < EE) ? evec[e * 3 + c] : 0.f;
  }
  for (int t = lane; t < 16 * 3; t += 32) {  // zero pad cols 53..55
    int r = t / 3, c = t - r * 3;
    ef[r * EDIM_PAD + EDIM + c] = 0.f;
  }
  __syncthreads();
  v8f acc0 = {}, acc1 = {};
  for (int k0 = 0; k0 < EDIM_PAD; k0 += 4) {
    v2f a = afrag(ef, EDIM_PAD, lane, k0);
    acc0 = wmma4(a, bfragP(wels, lane, k0, 0), acc0);
    acc1 = wmma4(a, bfragP(wels, lane, k0, 1), acc1);
  }
  int cb = lane & 15, rb = (lane >> 4) << 3;
  for (int r = 0; r < 8; ++r) {
    int row = rb + r;
    et[row * DD + cb] = acc0[r];
    et[row * DD + 16 + cb] = acc1[r];
  }
  __syncthreads();
}

// ---------------- edge pass 1: alpha + segment max ----------------
__global__ void k_edge_alpha(const float* __restrict__ elen,
                             const float* __restrict__ evec,
                             const int* __restrict__ eidx,
                             const float* __restrict__ We,
                             const float* __restrict__ qbuf,
                             const float* __restrict__ kbuf,
                             float* __restrict__ alpha,
                             unsigned* __restrict__ amax) {
  __shared__ float wels[EDIM_PAD * DD];
  __shared__ float efls[4][16 * EDIM_PAD];
  __shared__ float els[4][16 * DD];
  int wave = threadIdx.x >> 5, lane = threadIdx.x & 31;
  int base = (blockIdx.x * 4 + wave) * 16;
  stageW(We, wels, EDIM, EDIM_PAD);
  edge_gemm_tile(elen, evec, wels, efls[wave], els[wave], lane, base);
  const float* ew = els[wave];
  for (int t = lane; t < 16 * HH; t += 32) {  // (edge, head) tasks
    int m = t >> 2, hh = t & 3;
    int e = base + m;
    if (e < EE) {
      int src = eidx[e];
      int dst = eidx[EE + e];
      float4 qa = *(const float4*)(qbuf + dst * DD + hh * CC);
      float4 qb4 = *(const float4*)(qbuf + dst * DD + hh * CC + 4);
      float4 ka = *(const float4*)(kbuf + src * DD + hh * CC);
      float4 kb4 = *(const float4*)(kbuf + src * DD + hh * CC + 4);
      float4 ea = *(const float4*)(ew + m * DD + hh * CC);
      float4 eb = *(const float4*)(ew + m * DD + hh * CC + 4);
      float a = qa.x * (ka.x + ea.x) + qa.y * (ka.y + ea.y) +
                qa.z * (ka.z + ea.z) + qa.w * (ka.w + ea.w) +
                qb4.x * (kb4.x + eb.x) + qb4.y * (kb4.y + eb.y) +
                qb4.z * (kb4.z + eb.z) + qb4.w * (kb4.w + eb.w);
      a *= 0.35355339059327373f;  // 1/sqrt(8)
      alpha[e * HH + hh] = a;
      atomicMax(&amax[dst * HH + hh], fkey(a));
    }
  }
}

// ---------------- exp + segment sum ----------------
__global__ void k_exp(float* __restrict__ alpha, const int* __restrict__ eidx,
                      const unsigned* __restrict__ amax,
                      float* __restrict__ denom) {
  long long t = (long long)blockIdx.x * blockDim.x + threadIdx.x;
  if (t < (long long)EE * HH) {
    int e = (int)(t >> 2);
    int hh = (int)(t & 3);
    int dst = eidx[EE + e];
    float m = fdec(amax[dst * HH + hh]);
    float ex = __expf(alpha[t] - m);
    alpha[t] = ex;
    atomicAdd(&denom[dst * HH + hh], ex);
  }
}

// ---------------- edge pass 2: weighted messages ----------------
__global__ void k_edge_msg(const float* __restrict__ elen,
                           const float* __restrict__ evec,
                           const int* __restrict__ eidx,
                           const float* __restrict__ We,
                           const float* __restrict__ vbuf,
                           const float* __restrict__ alpha,
                           const float* __restrict__ denom,
                           float* __restrict__ agg) {
  __shared__ float wels[EDIM_PAD * DD];
  __shared__ float efls[4][16 * EDIM_PAD];
  __shared__ float els[4][16 * DD];
  __shared__ float wls[4][16 * HH];
  __shared__ int srcl[4][16];
  __shared__ int dstl[4][16];
  int wave = threadIdx.x >> 5, lane = threadIdx.x & 31;
  int base = (blockIdx.x * 4 + wave) * 16;
  stageW(We, wels, EDIM, EDIM_PAD);
  edge_gemm_tile(elen, evec, wels, efls[wave], els[wave], lane, base);
  for (int t = lane; t < 16 * HH; t += 32) {  // (edge, head) tasks
    int m = t >> 2, hh = t & 3;
    int e = base + m;
    if (e < EE) {
      int src = eidx[e];
      int dst = eidx[EE + e];
      if (hh == 0) {
        srcl[wave][m] = src;
        dstl[wave][m] = dst;
      }
      float ex = alpha[e * HH + hh];
      wls[wave][m * HH + hh] = ex / (denom[dst * HH + hh] + 1e-16f);
    } else if (hh == 0) {
      dstl[wave][m] = -1;
    }
  }
  __syncthreads();
  const float* ew = els[wave];
  for (int idx = lane; idx < 16 * 8; idx += 32) {  // (edge, float4-chunk)
    int m = idx >> 3, c4 = idx & 7;
    int dst = dstl[wave][m];
    if (dst >= 0) {
      float4 vv = *(const float4*)(vbuf + srcl[wave][m] * DD + c4 * 4);
      float4 ee = *(const float4*)(ew + m * DD + c4 * 4);
      float wv = wls[wave][m * HH + (c4 >> 1)];
      float* a = agg + dst * DD + c4 * 4;
      atomicAdd(a + 0, (vv.x + ee.x) * wv);
      atomicAdd(a + 1, (vv.y + ee.y) * wv);
      atomicAdd(a + 2, (vv.z + ee.z) * wv);
      atomicAdd(a + 3, (vv.w + ee.w) * wv);
    }
  }
}

// ---------------- residual update ----------------
__global__ void k_residual(float* __restrict__ h, const float* __restrict__ agg,
                           const float* __restrict__ sk,
                           float* __restrict__ out) {
  int t = blockIdx.x * blockDim.x + threadIdx.x;
  if (t < NN * DD) {
    float r = h[t] + agg[t] + sk[t];
    h[t] = r;
    out[t] = r;
  }
}

// ---------------- per-layer reset ----------------
__global__ void k_init(float* __restrict__ agg, float* __restrict__ denom,
                       unsigned* __restrict__ amax) {
  int t = blockIdx.x * blockDim.x + threadIdx.x;
  if (t < NN * DD) agg[t] = 0.f;
  if (t < NN * HH) {
    denom[t] = 0.f;
    amax[t] = 0u;  // encodes -inf under fkey ordering
  }
}

extern "C" void kernel_launch(void* const* d_in, const int* in_sizes, int n_in,
                              void* d_out, int out_size, void* d_ws,
                              size_t ws_size, hipStream_t stream) {
  const float* x = (const float*)d_in[0];
  const int* eidx = (const int*)d_in[1];
  const float* elen = (const float*)d_in[2];
  const float* evec = (const float*)d_in[3];
  const float* niW = (const float*)d_in[4];
  const float* nib = (const float*)d_in[5];
  const float* lng = (const float*)d_in[6];
  const float* lnb = (const float*)d_in[7];
  const float* Wq = (const float*)d_in[8];
  const float* bq = (const float*)d_in[9];
  const float* Wk = (const float*)d_in[10];
  const float* bk = (const float*)d_in[11];
  const float* Wv = (const float*)d_in[12];
  const float* bv = (const float*)d_in[13];
  const float* We = (const float*)d_in[14];
  const float* Ws = (const float*)d_in[15];
  const float* bs = (const float*)d_in[16];

  float* w = (float*)d_ws;
  float* h = w;        w += (size_t)NN * DD;
  float* qb = w;       w += (size_t)NN * DD;
  float* kb = w;       w += (size_t)NN * DD;
  float* vb = w;       w += (size_t)NN * DD;
  float* sk = w;       w += (size_t)NN * DD;
  float* agg = w;      w += (size_t)NN * DD;
  float* denom = w;    w += (size_t)NN * HH;
  unsigned* amax = (unsigned*)w; w += (size_t)NN * HH;
  float* alpha = w;    w += (size_t)EE * HH;

  int nodeTiles = (NN + 15) / 16;
  int nodeBlocks = (nodeTiles + 3) / 4;
  int edgeTiles = (EE + 15) / 16;
  int edgeBlocks = (edgeTiles + 3) / 4;
  int elemBlocks = (NN * DD + 255) / 256;
  long long ehTotal = (long long)EE * HH;
  int expBlocks = (int)((ehTotal + 255) / 256);

  k_input_proj<<<nodeBlocks, 128, 0, stream>>>(x, niW, nib, h);
  for (int l = 0; l < LL; ++l) {
    k_init<<<elemBlocks, 256, 0, stream>>>(agg, denom, amax);
    k_ln_proj<<<nodeBlocks, 128, 0, stream>>>(
        h, lng + l * DD, lnb + l * DD, Wq + l * DD * DD, bq + l * DD,
        Wk + l * DD * DD, bk + l * DD, Wv + l * DD * DD, bv + l * DD,
        Ws + l * DD * DD, bs + l * DD, qb, kb, vb, sk);
    k_edge_alpha<<<edgeBlocks, 128, 0, stream>>>(elen, evec, eidx,
                                                 We + l * EDIM * DD, qb, kb,
                                                 alpha, amax);
    k_exp<<<expBlocks, 256, 0, stream>>>(alpha, eidx, amax, denom);
    k_edge_msg<<<edgeBlocks, 128, 0, stream>>>(elen, evec, eidx,
                                               We + l * EDIM * DD, vb, alpha,
                                               denom, agg);
    k_residual<<<elemBlocks, 256, 0, stream>>>(
        h, agg, sk, (l == LL - 1) ? (float*)d_out : h);
  }
}